// IntegratedTGNLayer_32770600468497
// MI455X (gfx1250) — compile-verified
//
#include <hip/hip_runtime.h>
#include <math.h>

// ---------------- types / constants ----------------
typedef float v2f __attribute__((ext_vector_type(2)));
typedef float v4f __attribute__((ext_vector_type(4)));
typedef float v8f __attribute__((ext_vector_type(8)));

#define B_      131072
#define TILE_M  64
#define XS      516          // padded row stride of X tile (bank-conflict free)
#define HSTR    116          // padded row stride of hid/msg tiles
#define HID0    (TILE_M*XS)              // 33024 floats
#define HSZ     (TILE_M*HSTR)            // 7424 floats
#define MSG0    (HID0 + 2*HSZ)
#define LDS_FLOATS (MSG0 + 2*HSZ)        // 62720 floats = 250880 B  (<320KB)

#define WMMA4(A,Bv,C) __builtin_amdgcn_wmma_f32_16x16x4_f32(false,(A),false,(Bv),(short)0,(C),false,false)

__device__ __forceinline__ float sigmoidf_(float x) { return 1.0f / (1.0f + __expf(-x)); }

// ---- weight prep: repack all weights into WMMA-fragment order  F[k/4][n][kk] = W[k][n]
// so a B fragment is ONE b64 load:  *(v2f*)(F + (k>>2)*4N + n*4 + kb)
__global__ void tgn_prep(const float* __restrict__ w1, const float* __restrict__ w2,
                         const float* __restrict__ wih, const float* __restrict__ whh,
                         const float* __restrict__ ow,
                         float* __restrict__ w1f, float* __restrict__ w2f,
                         float* __restrict__ wihTf, float* __restrict__ whhTf,
                         float* __restrict__ owf) {
    const int N1 = 121*448;          // w1  : K=484, N=112 (padded from 100)
    const int N2 = 25*448;           // w2  : K=100, N=112
    const int N3 = 25*1536;          // wih^T: K=100, N=384
    const int N4 = 32*1536;          // whh^T: K=128, N=384
    const int N5 = 64*512;           // out_w: K=256, N=128
    int i = blockIdx.x * blockDim.x + threadIdx.x;
    int stride = gridDim.x * blockDim.x;
    for (int idx = i; idx < N1 + N2 + N3 + N4 + N5; idx += stride) {
        if (idx < N1) {
            int q = idx / 448, r = idx % 448, n = r >> 2, kk = r & 3, k = 4*q + kk;
            w1f[idx] = (n < 100) ? w1[k*100 + n] : 0.0f;
        } else if (idx < N1 + N2) {
            int j = idx - N1, q = j / 448, r = j % 448, n = r >> 2, kk = r & 3, k = 4*q + kk;
            w2f[j] = (n < 100) ? w2[k*100 + n] : 0.0f;
        } else if (idx < N1 + N2 + N3) {
            int j = idx - N1 - N2, q = j / 1536, r = j % 1536, n = r >> 2, kk = r & 3, k = 4*q + kk;
            wihTf[j] = wih[n*100 + k];           // transpose on the fly
        } else if (idx < N1 + N2 + N3 + N4) {
            int j = idx - N1 - N2 - N3, q = j / 1536, r = j % 1536, n = r >> 2, kk = r & 3, k = 4*q + kk;
            whhTf[j] = whh[n*128 + k];
        } else {
            int j = idx - N1 - N2 - N3 - N4, q = j / 512, r = j % 512, n = r >> 2, kk = r & 3, k = 4*q + kk;
            owf[j] = ow[k*128 + n];
        }
    }
}

// ---------------- bulk copy memory -> new_memory (512MB, NT stores keep L2 clean) --------
__global__ void tgn_copy(const v4f* __restrict__ s, v4f* __restrict__ d, int n) {
    int i = blockIdx.x * blockDim.x + threadIdx.x;
    int stride = gridDim.x * blockDim.x;
    for (; i < n; i += stride) {
        v4f v = s[i];
        __builtin_nontemporal_store(v, d + i);
    }
}

// ---------------- scatter updated rows into new_memory (8 rows / block) ----------------
__global__ void tgn_scatter(const int* __restrict__ ids, const float* __restrict__ upd,
                            float* __restrict__ nm) {
    int r = blockIdx.x * 8 + (threadIdx.x >> 5);
    int c = (threadIdx.x & 31) << 2;
    int id = ids[r];
    v4f v = *(const v4f*)(upd + (size_t)r*128 + c);
    __builtin_nontemporal_store(v, (v4f*)(nm + (size_t)id*128 + c));
}

// ---------------- fused TGN layer: gather + time-enc + MLP + GRU + out-proj ----------------
__global__ __launch_bounds__(256, 1)
void tgn_fused(const float* __restrict__ mem,
               const int*   __restrict__ sid,  const int*   __restrict__ did,
               const float* __restrict__ edge, const float* __restrict__ ts,
               const float* __restrict__ tw,   const float* __restrict__ tb,
               const float* __restrict__ w1f,  const float* __restrict__ b1,
               const float* __restrict__ w2f,  const float* __restrict__ b2,
               const float* __restrict__ wihTf, const float* __restrict__ bih,
               const float* __restrict__ whhTf, const float* __restrict__ bhh,
               const float* __restrict__ owf,  const float* __restrict__ outb,
               const float* __restrict__ semb, const float* __restrict__ demb,
               float* __restrict__ updS, float* __restrict__ updD,
               float* __restrict__ out) {
    extern __shared__ float lds[];
    const int tid  = threadIdx.x;
    const int row0 = blockIdx.x * TILE_M;
    const int lane = tid & 31;
    const int l16  = lane & 15;
    const int kb   = (lane >> 4) << 1;   // K sub-offset of A/B fragments: 0 or 2
    const int m0l  = (lane >> 4) << 3;   // M sub-offset of C fragments: 0 or 8
    const int wv   = tid >> 5;           // wave id 0..7
    const v8f Z = {0.f,0.f,0.f,0.f,0.f,0.f,0.f,0.f};

    // ---- phase 0: stage X = [src_mem | dst_mem | edge | time_emb] into LDS ----
    for (int idx = tid; idx < TILE_M*32; idx += 256) {
        int r = idx >> 5, c4 = (idx & 31) << 2;
        int ns = sid[row0 + r];
        v4f v = *(const v4f*)(mem + (size_t)ns*128 + c4);
        *(v4f*)(&lds[r*XS + c4]) = v;
    }
    for (int idx = tid; idx < TILE_M*32; idx += 256) {
        int r = idx >> 5, c4 = (idx & 31) << 2;
        int nd = did[row0 + r];
        v4f v = *(const v4f*)(mem + (size_t)nd*128 + c4);
        *(v4f*)(&lds[r*XS + 128 + c4]) = v;
    }
    for (int idx = tid; idx < TILE_M*32; idx += 256) {
        int r = idx >> 5, c4 = (idx & 31) << 2;
        v4f v = *(const v4f*)(edge + (size_t)(row0 + r)*128 + c4);
        *(v4f*)(&lds[r*XS + 256 + c4]) = v;
    }
    for (int idx = tid; idx < TILE_M*100; idx += 256) {
        int r = idx / 100, j = idx - r*100;
        lds[r*XS + 384 + j] = cosf(ts[row0 + r] * tw[j] + tb[j]);
    }
    __syncthreads();

    // ---- phase 1: GEMM1  relu(X @ w1 + b1) -> hid[v]   (K=484, N=112) ----
    // unit = (v, nt); one B fragment feeds 4 M-tiles.  d2s = src/dst column swap,
    // done with 3 segment loops (uniform A offset, no per-iter select).
    for (int unit = wv; unit < 14; unit += 8) {
        const int v  = unit / 7, nt = unit % 7;
        const int n    = nt*16 + l16;
        const int n4kb = n*4 + kb;
        v8f acc[4];
#pragma unroll
        for (int mt = 0; mt < 4; ++mt) acc[mt] = Z;

#define G1_STEP(ACOL)                                                          \
        {                                                                      \
            v2f b = *(const v2f*)(w1f + ((k >> 2) * 448) + n4kb);              \
            const int ac_ = (ACOL) + kb;                                       \
            _Pragma("unroll")                                                  \
            for (int mt = 0; mt < 4; ++mt) {                                   \
                v2f a = *(const v2f*)(&lds[(mt*16 + l16)*XS + ac_]);           \
                acc[mt] = WMMA4(a, b, acc[mt]);                                \
            }                                                                  \
        }
        const int sw = v ? 128 : 0;
        int k = 0;
        for (; k < 128; k += 4) G1_STEP(k + sw)
        for (; k < 256; k += 4) G1_STEP(k - sw)
        for (; k < 484; k += 4) G1_STEP(k)
#undef G1_STEP

        const float bias = (n < 100) ? b1[n] : 0.0f;
        float* H = &lds[HID0 + v*HSZ];
#pragma unroll
        for (int mt = 0; mt < 4; ++mt) {
            const int m0 = mt*16 + m0l;
#pragma unroll
            for (int i = 0; i < 8; ++i) {
                float x = acc[mt][i] + bias;
                H[(m0 + i)*HSTR + n] = x > 0.f ? x : 0.f;
            }
        }
    }
    __syncthreads();

    // ---- phase 2: GEMM2  hid @ w2 + b2 -> msg[v]   (K=100, N=112) ----
    for (int unit = wv; unit < 14; unit += 8) {
        const int v  = unit / 7, nt = unit % 7;
        const int n    = nt*16 + l16;
        const int n4kb = n*4 + kb;
        const float* H = &lds[HID0 + v*HSZ];
        v8f acc[4];
#pragma unroll
        for (int mt = 0; mt < 4; ++mt) acc[mt] = Z;
        for (int k = 0; k < 100; k += 4) {
            v2f b = *(const v2f*)(w2f + ((k >> 2) * 448) + n4kb);
#pragma unroll
            for (int mt = 0; mt < 4; ++mt) {
                v2f a = *(const v2f*)(H + (mt*16 + l16)*HSTR + k + kb);
                acc[mt] = WMMA4(a, b, acc[mt]);
            }
        }
        const float bias = (n < 100) ? b2[n] : 0.0f;
        float* M = &lds[MSG0 + v*HSZ];
#pragma unroll
        for (int mt = 0; mt < 4; ++mt) {
            const int m0 = mt*16 + m0l;
#pragma unroll
            for (int i = 0; i < 8; ++i) M[(m0 + i)*HSTR + n] = acc[mt][i] + bias;
        }
    }
    __syncthreads();

    // ---- phases 3..6: GRU update + output projection, src (u=0) then dst (u=1) ----
    for (int u = 0; u < 2; ++u) {
        const int mv   = u ^ 1;                 // src update consumes d2s msg, dst consumes s2d
        const int hoff = u ? 128 : 0;           // h = Xs or Xd columns of the X tile
        const float* emb  = u ? demb : semb;
        float*       updg = u ? updD : updS;

        // stage node embeddings into X cols 384..511 (time cols are dead now)
        for (int idx = tid; idx < TILE_M*32; idx += 256) {
            int r = idx >> 5, c4 = (idx & 31) << 2;
            v4f v = *(const v4f*)(emb + (size_t)(row0 + r)*128 + c4);
            *(v4f*)(&lds[r*XS + 384 + c4]) = v;
        }

        // GRU: wave wv owns column tile nt=wv (j = 128 cols / 8 waves).
        // 2 M-tiles at a time: 12 live accumulators, 3 B fragments feed 6 WMMAs.
        const int j    = wv*16 + l16;           // 0..127
        const int j4kb = j*4 + kb;
        const float* M = &lds[MSG0 + mv*HSZ];
        const float bir = bih[j], biz = bih[128 + j], bin_ = bih[256 + j];
        const float bhr = bhh[j], bhz = bhh[128 + j], bhn  = bhh[256 + j];

#pragma unroll
        for (int mh = 0; mh < 2; ++mh) {
            v8f gr[2], gz[2], gn[2], hr[2], hz[2], hn[2];
#pragma unroll
            for (int m = 0; m < 2; ++m) { gr[m]=Z; gz[m]=Z; gn[m]=Z; hr[m]=Z; hz[m]=Z; hn[m]=Z; }

            for (int k = 0; k < 100; k += 4) {          // gi = msg @ w_ih^T, K=100
                const float* Bq = wihTf + ((k >> 2) * 1536);
                v2f br_ = *(const v2f*)(Bq + j4kb);
                v2f bz_ = *(const v2f*)(Bq + 512 + j4kb);
                v2f bn_ = *(const v2f*)(Bq + 1024 + j4kb);
#pragma unroll
                for (int m = 0; m < 2; ++m) {
                    v2f a = *(const v2f*)(M + ((mh*2 + m)*16 + l16)*HSTR + k + kb);
                    gr[m] = WMMA4(a, br_, gr[m]);
                    gz[m] = WMMA4(a, bz_, gz[m]);
                    gn[m] = WMMA4(a, bn_, gn[m]);
                }
            }
            for (int k = 0; k < 128; k += 4) {          // gh = h @ w_hh^T, K=128
                const float* Bq = whhTf + ((k >> 2) * 1536);
                v2f br_ = *(const v2f*)(Bq + j4kb);
                v2f bz_ = *(const v2f*)(Bq + 512 + j4kb);
                v2f bn_ = *(const v2f*)(Bq + 1024 + j4kb);
#pragma unroll
                for (int m = 0; m < 2; ++m) {
                    v2f a = *(const v2f*)(&lds[((mh*2 + m)*16 + l16)*XS + hoff + k + kb]);
                    hr[m] = WMMA4(a, br_, hr[m]);
                    hz[m] = WMMA4(a, bz_, hz[m]);
                    hn[m] = WMMA4(a, bn_, hn[m]);
                }
            }
#pragma unroll
            for (int m = 0; m < 2; ++m) {
                const int m0 = (mh*2 + m)*16 + m0l;
#pragma unroll
                for (int i = 0; i < 8; ++i) {
                    float rg = sigmoidf_(gr[m][i] + bir + hr[m][i] + bhr);
                    float zg = sigmoidf_(gz[m][i] + biz + hz[m][i] + bhz);
                    float ng = tanhf(gn[m][i] + bin_ + rg * (hn[m][i] + bhn));
                    float hv = lds[(m0 + i)*XS + hoff + j];
                    float uu = (1.0f - zg) * ng + zg * hv;
                    lds[(m0 + i)*XS + 256 + j] = uu;                   // for out-proj concat
                    updg[(size_t)(row0 + m0 + i)*128 + j] = uu;        // staging for scatter
                }
            }
        }
        __syncthreads();

        // out = [upd | emb] @ out_w + out_b   (A = X cols 256..511, K=256; wave owns nt=wv)
        const size_t obase = (size_t)(u ? (B_ + row0) : row0);
        {
            const int n    = wv*16 + l16;
            const int n4kb = n*4 + kb;
            v8f acc[4];
#pragma unroll
            for (int mt = 0; mt < 4; ++mt) acc[mt] = Z;
            for (int k = 0; k < 256; k += 4) {
                v2f b = *(const v2f*)(owf + ((k >> 2) * 512) + n4kb);
#pragma unroll
                for (int mt = 0; mt < 4; ++mt) {
                    v2f a = *(const v2f*)(&lds[(mt*16 + l16)*XS + 256 + k + kb]);
                    acc[mt] = WMMA4(a, b, acc[mt]);
                }
            }
            const float bias = outb[n];
#pragma unroll
            for (int mt = 0; mt < 4; ++mt) {
                const int m0 = mt*16 + m0l;
#pragma unroll
                for (int i = 0; i < 8; ++i)
                    out[(obase + m0 + i)*128 + n] = acc[mt][i] + bias;
            }
        }
        __syncthreads();   // cols 256..511 reused by the u=1 pass
    }
}

// ---------------- launch ----------------
extern "C" void kernel_launch(void* const* d_in, const int* in_sizes, int n_in,
                              void* d_out, int out_size, void* d_ws, size_t ws_size,
                              hipStream_t stream) {
    (void)in_sizes; (void)n_in; (void)out_size; (void)ws_size;
    const float* semb = (const float*)d_in[0];
    const float* demb = (const float*)d_in[1];
    const float* edge = (const float*)d_in[2];
    const float* ts   = (const float*)d_in[3];
    const float* mem  = (const float*)d_in[4];
    const float* tw   = (const float*)d_in[5];
    const float* tb   = (const float*)d_in[6];
    const float* w1   = (const float*)d_in[7];
    const float* b1   = (const float*)d_in[8];
    const float* w2   = (const float*)d_in[9];
    const float* b2   = (const float*)d_in[10];
    const float* wih  = (const float*)d_in[11];
    const float* bih  = (const float*)d_in[12];
    const float* whh  = (const float*)d_in[13];
    const float* bhh  = (const float*)d_in[14];
    const float* outw = (const float*)d_in[15];
    const float* outb = (const float*)d_in[16];
    const int*   sid  = (const int*)d_in[17];
    const int*   did  = (const int*)d_in[18];

    float* out    = (float*)d_out;
    float* newmem = out + (size_t)2 * B_ * 128;      // new_memory region of d_out

    float* ws    = (float*)d_ws;
    float* w1f   = ws;                               // 121*448 = 54208
    float* w2f   = ws + 54272;                       // 25*448  = 11200
    float* wihTf = ws + 65536;                       // 25*1536 = 38400
    float* whhTf = ws + 104448;                      // 32*1536 = 49152
    float* owf   = ws + 153600;                      // 64*512  = 32768
    float* updS  = ws + 186368;                      // B*128
    float* updD  = updS + (size_t)B_ * 128;          // B*128

    tgn_prep<<<726, 256, 0, stream>>>(w1, w2, wih, whh, outw, w1f, w2f, wihTf, whhTf, owf);

    // copy memory -> new_memory (1M x 128 f32 = 512MB, pure bandwidth)
    tgn_copy<<<8192, 256, 0, stream>>>((const v4f*)mem, (v4f*)newmem, 32000000);

    tgn_fused<<<B_/TILE_M, 256, LDS_FLOATS*sizeof(float), stream>>>(
        mem, sid, did, edge, ts, tw, tb,
        w1f, b1, w2f, b2, wihTf, bih, whhTf, bhh, owf, outb,
        semb, demb, updS, updD, out);

    // reference order: scatter src rows first, then dst rows (dst wins on collision)
    tgn_scatter<<<B_/8, 256, 0, stream>>>(sid, updS, newmem);
    tgn_scatter<<<B_/8, 256, 0, stream>>>(did, updD, newmem);
}